// GPTQLinear_39341900431687
// MI455X (gfx1250) — compile-verified
//
#include <hip/hip_runtime.h>

typedef __attribute__((ext_vector_type(2))) float v2f;
typedef __attribute__((ext_vector_type(8))) float v8f;

#define IN_F   8192
#define OUT_F  8192
#define BATCH  16
#define WAVES  4
#define KSPAN  (IN_F / WAVES)   // 2048 K per wave
#define KSTEP  32               // 4 b128 pairs -> 8 WMMAs per iteration

// One block = one 16-wide N tile (16x16 C tile), 4 waves split K=8192.
// Memory-bound: per lane, one global_load_b128 of int32 weights + one of
// (L2-hot) x feeds TWO V_WMMA_F32_16X16X4_F32 ops. K-slots are permuted
// (half h holds k0+4h..k0+4h+3) identically for A and B, which is legal
// since the K-contraction is order-independent. Double-buffered so 8 b128
// loads stay in flight during each WMMA block.
__global__ __launch_bounds__(128) void gptq_wmma_f32_kernel(
    const float* __restrict__ x,       // [BATCH, IN_F]
    const int*   __restrict__ qw,      // [OUT_F, IN_F], values in [-8,7]
    const float* __restrict__ scale_p, // scalar
    const float* __restrict__ bias,    // [OUT_F]
    float*       __restrict__ out)     // [BATCH, OUT_F]
{
    __shared__ float lds[WAVES * 8 * 32];   // 4 KB of partial C tiles

    const int tid  = threadIdx.x;
    const int wave = tid >> 5;
    const int lane = tid & 31;
    const int half = lane >> 4;   // which K-quad this lane-half streams
    const int nl   = lane & 15;   // M for A, N for B
    const int n0   = blockIdx.x * 16;

    const int kbeg = wave * KSPAN;

    // Per-lane base pointers; half h covers logical k = kc + 4h .. kc + 4h+3.
    const float* xp = x  + nl * IN_F + kbeg + 4 * half;
    const int*   wp = qw + (size_t)(n0 + nl) * IN_F + kbeg + 4 * half;

    v8f c = {};  // 16x16 f32 accumulator (8 VGPRs / lane)

    // Prime the pipeline: chunk at k = 0.
    float4 xb[4];
    int4   wb[4];
    #pragma unroll
    for (int j = 0; j < 4; ++j) {
        xb[j] = *(const float4*)(xp + 8 * j);
        wb[j] = *(const int4*)  (wp + 8 * j);
    }

    for (int k = 0; k < KSPAN; k += KSTEP) {
        // Issue next chunk's loads first (wraps to 0 on the last iteration:
        // valid address, result unused; keeps control flow uniform).
        const int knext = (k + KSTEP) & (KSPAN - 1);
        __builtin_prefetch(wp + k + 1024, 0, 1);   // ~4KB ahead per lane
        float4 xn[4];
        int4   wn[4];
        #pragma unroll
        for (int j = 0; j < 4; ++j) {
            xn[j] = *(const float4*)(xp + knext + 8 * j);
            wn[j] = *(const int4*)  (wp + knext + 8 * j);
        }

        // Consume current chunk: 8 WMMAs. Only waits on the 8 older loads.
        #pragma unroll
        for (int j = 0; j < 4; ++j) {
            v2f a0, b0, a1, b1;
            a0.x = xb[j].x;        a0.y = xb[j].y;
            b0.x = (float)wb[j].x; b0.y = (float)wb[j].y;
            c = __builtin_amdgcn_wmma_f32_16x16x4_f32(
                    false, a0, false, b0, (short)0, c, false, false);
            a1.x = xb[j].z;        a1.y = xb[j].w;
            b1.x = (float)wb[j].z; b1.y = (float)wb[j].w;
            c = __builtin_amdgcn_wmma_f32_16x16x4_f32(
                    false, a1, false, b1, (short)0, c, false, false);
        }

        // Rotate buffers.
        #pragma unroll
        for (int j = 0; j < 4; ++j) {
            xb[j] = xn[j];
            wb[j] = wn[j];
        }
    }

    // Stash this wave's partial C tile to LDS.
    #pragma unroll
    for (int v = 0; v < 8; ++v)
        lds[(wave * 8 + v) * 32 + lane] = c[v];
    __syncthreads();

    // Cross-wave reduction + epilogue: 256 (lane, vgpr) entries / 128 threads.
    const float s = *scale_p;
    #pragma unroll
    for (int r = 0; r < 2; ++r) {
        const int e = tid + r * 128;
        const int l = e & 31;
        const int v = e >> 5;
        float sum = 0.f;
        #pragma unroll
        for (int w = 0; w < WAVES; ++w)
            sum += lds[(w * 8 + v) * 32 + l];
        // C/D layout: VGPR v, lanes 0-15 -> M=v, lanes 16-31 -> M=v+8; N=lane%16.
        const int m = v + 8 * (l >> 4);
        const int n = n0 + (l & 15);
        out[m * OUT_F + n] = sum * s + bias[n];
    }
}

extern "C" void kernel_launch(void* const* d_in, const int* in_sizes, int n_in,
                              void* d_out, int out_size, void* d_ws, size_t ws_size,
                              hipStream_t stream) {
    const float* x     = (const float*)d_in[0];  // [16, 8192] f32
    const int*   qw    = (const int*)  d_in[1];  // [8192, 8192] i32
    const float* scale = (const float*)d_in[2];  // scalar f32
    const float* bias  = (const float*)d_in[3];  // [8192] f32
    float*       out   = (float*)d_out;          // [16, 8192] f32

    (void)in_sizes; (void)n_in; (void)out_size; (void)d_ws; (void)ws_size;

    dim3 grid(OUT_F / 16);   // 512 blocks, one 16-wide N tile each
    dim3 block(128);         // 4 waves split K
    gptq_wmma_f32_kernel<<<grid, block, 0, stream>>>(x, qw, scale, bias, out);
}